// D3PMLVBLoss_76708115906767
// MI455X (gfx1250) — compile-verified
//
#include <hip/hip_runtime.h>

// D3PM LVB loss for B=64, L=2048, K=30, V=33, TMAX=500 on gfx1250.
// Stage 1: per-(block) partial masked-loss + masked-count, with the
//          (p*p) @ Q_bar[t-1] matvec done via V_WMMA_F32_16X16X4_F32.
// Stage 2: deterministic per-batch normalization + mean.

#define B_    64
#define L_    2048
#define K_    30
#define V_    33
#define TMAX_ 500
#define KP    32          // K padded to 32 for WMMA
#define TS    33          // LDS tile row stride (odd -> conflict-free)
#define TILE_L 256
#define NBLK  (L_ / TILE_L)   // 8 L-tiles per batch row

typedef __attribute__((ext_vector_type(2))) float v2f;
typedef __attribute__((ext_vector_type(8))) float v8f;

__global__ __launch_bounds__(256) void d3pm_stage1(
    const float* __restrict__ src_onehot,   // [B,L,K]
    const float* __restrict__ qin,          // [B,L,K]
    const float* __restrict__ predictions,  // [B,L,V]
    const int*   __restrict__ tgt,          // [B,L]
    const float* __restrict__ input_mask,   // [B,L]
    const int*   __restrict__ timesteps,    // [B]
    const float* __restrict__ Qmat,         // [TMAX+1,K,K]
    const float* __restrict__ Qbar,         // [TMAX+1,K,K]
    float* __restrict__ partLoss,           // [B,NBLK]
    float* __restrict__ partMask)           // [B,NBLK]
{
    __shared__ float sQtT[KP * TS];          // sQtT[x][k] = Qt[k][x]
    __shared__ float sQbm[KP * TS];          // sQbm[j][k] = Qbar[t-1][j][k]
    __shared__ float sTiles[8 * KP * TS];    // per-wave 32x32 pp tile (reused for s)
    __shared__ float sRedV[256];
    __shared__ float sRedM[256];

    const int b    = blockIdx.y;
    const int tid  = threadIdx.x;
    const int lane = tid & 31;
    const int wave = tid >> 5;
    const int l    = blockIdx.x * TILE_L + tid;       // one position per thread
    const int t    = timesteps[b];                    // uniform per block

    const size_t pos = (size_t)b * L_ + (size_t)l;
    const float msk  = (input_mask[pos] > 0.5f) ? 1.0f : 0.0f;

    float val = 0.0f;

    if (t == 1) {
        // ---- cross-entropy branch: -log_softmax(pred[:, :30])[tgt]
        const float* row = predictions + pos * V_;
        const int tg = tgt[pos];
        float xv[K_];
        float mx = -3.0e38f, xtg = 0.0f;
#pragma unroll
        for (int k = 0; k < K_; ++k) {
            const float x = row[k];
            xv[k] = x;
            mx = fmaxf(mx, x);
            if (k == tg) xtg = x;
        }
        float Z = 0.0f;
#pragma unroll
        for (int k = 0; k < K_; ++k) Z += __expf(xv[k] - mx);
        val = mx + __logf(Z) - xtg;                   // ce_pos
    } else if (t == TMAX_) {
        // ---- prior-KL branch: sum q_hat * (log q_hat + log K)
        const float* row = qin + pos * K_;
        float S = 0.0f, acc = 0.0f;
#pragma unroll
        for (int k = 0; k < K_; ++k) {
            const float v = row[k];
            S   += v;
            acc += v * __logf(v);
        }
        val = acc / S - __logf(S) + __logf((float)K_);
    } else {
        // ---- generic KL branch
        const float* Qt = Qmat + (size_t)t * (K_ * K_);
        const float* Qm = Qbar + (size_t)(t - 1) * (K_ * K_);
        // stage Qt^T and Qbar[t-1] into zero-padded 32x32 LDS tiles
        for (int i = tid; i < KP * KP; i += 256) {
            const int r = i >> 5, c = i & (KP - 1);
            sQtT[r * TS + c] = (r < K_ && c < K_) ? Qt[c * K_ + r] : 0.0f; // [x][k]
            sQbm[r * TS + c] = (r < K_ && c < K_) ? Qm[r * K_ + c] : 0.0f; // [j][k]
        }
        __syncthreads();   // uniform: whole block is in this branch

        // softmax over first 30 logits -> p, write p^2 row to per-wave tile
        const float* row = predictions + pos * V_;
        float xv[K_];
        float mx = -3.0e38f;
#pragma unroll
        for (int k = 0; k < K_; ++k) { xv[k] = row[k]; mx = fmaxf(mx, xv[k]); }
        float Z = 0.0f;
#pragma unroll
        for (int k = 0; k < K_; ++k) { xv[k] = __expf(xv[k] - mx); Z += xv[k]; }
        const float invZ = 1.0f / Z;

        float* tile = sTiles + wave * (KP * TS);
#pragma unroll
        for (int k = 0; k < K_; ++k) {
            const float p = xv[k] * invZ;
            tile[lane * TS + k] = p * p;
        }
        tile[lane * TS + 30] = 0.0f;
        tile[lane * TS + 31] = 0.0f;

        // recover xt from the one-hot row; tgt comes in as int
        const float* srow = src_onehot + pos * K_;
        int xt = 0;
#pragma unroll
        for (int k = 0; k < K_; ++k) if (srow[k] > 0.5f) xt = k;
        const int tg = tgt[pos];

        // ---- s = (p*p) @ Qbar[t-1] via V_WMMA_F32_16X16X4_F32
        // wave covers 32 positions: 2 M-tiles x 2 N-tiles x 8 K-chunks of 4
        const int mrow = lane & 15;
        const int grp  = lane >> 4;      // 0: K+{0,1}, 1: K+{2,3} (f32 operand layout)
        v8f acc[2][2];
#pragma unroll
        for (int hm = 0; hm < 2; ++hm) {
#pragma unroll
            for (int nt = 0; nt < 2; ++nt) {
                v8f c = {0.f, 0.f, 0.f, 0.f, 0.f, 0.f, 0.f, 0.f};
#pragma unroll
                for (int ch = 0; ch < 8; ++ch) {
                    const int kb = 4 * ch + 2 * grp;
                    v2f av, bv;
                    av.x = tile[(hm * 16 + mrow) * TS + kb];
                    av.y = tile[(hm * 16 + mrow) * TS + kb + 1];
                    bv.x = sQbm[kb * TS + (nt * 16 + mrow)];
                    bv.y = sQbm[(kb + 1) * TS + (nt * 16 + mrow)];
                    c = __builtin_amdgcn_wmma_f32_16x16x4_f32(
                        false, av, false, bv, (short)0, c, false, false);
                }
                acc[hm][nt] = c;
            }
        }
        // scatter C/D layout (vgpr r -> rows r / r+8) back into the tile as s
#pragma unroll
        for (int hm = 0; hm < 2; ++hm)
#pragma unroll
            for (int nt = 0; nt < 2; ++nt)
#pragma unroll
                for (int r = 0; r < 8; ++r)
                    tile[(hm * 16 + r + 8 * grp) * TS + (nt * 16 + mrow)] = acc[hm][nt][r];

        // ---- finalize: KL(q_tm1 || p_theta); denom cancels, u/w = Bm/s
        float Su = 0.0f, Sw = 0.0f, a2 = 0.0f;
#pragma unroll
        for (int k = 0; k < K_; ++k) {
            const float a  = sQtT[xt * TS + k];       // Qt[k, xt]
            const float bm = sQbm[tg * TS + k];       // Qbar[t-1][tgt, k]
            const float sv = tile[lane * TS + k];     // s[k]
            const float u  = a * bm;
            const float w  = a * sv;
            Su += u;
            Sw += w;
            a2 += u * (__logf(bm) - __logf(sv));
        }
        val = a2 / Su + __logf(Sw) - __logf(Su);
    }

    // ---- deterministic block reduction of masked loss + mask count
    sRedV[tid] = val * msk;
    sRedM[tid] = msk;
    __syncthreads();
    for (int s = 128; s > 0; s >>= 1) {
        if (tid < s) { sRedV[tid] += sRedV[tid + s]; sRedM[tid] += sRedM[tid + s]; }
        __syncthreads();
    }
    if (tid == 0) {
        partLoss[b * NBLK + blockIdx.x] = sRedV[0];
        partMask[b * NBLK + blockIdx.x] = sRedM[0];
    }
}

__global__ __launch_bounds__(64) void d3pm_stage2(
    const float* __restrict__ partLoss,
    const float* __restrict__ partMask,
    float* __restrict__ out)
{
    __shared__ float r[64];
    const int tid = threadIdx.x;   // one thread per batch sample
    float loss = 0.0f, dlen = 0.0f;
#pragma unroll
    for (int i = 0; i < NBLK; ++i) {
        loss += partLoss[tid * NBLK + i];
        dlen += partMask[tid * NBLK + i];
    }
    float per = 0.0f;
    if (dlen > 0.0f) per = loss / fmaxf(dlen, 1.0f);
    r[tid] = per;
    __syncthreads();
    for (int s = 32; s > 0; s >>= 1) {
        if (tid < s) r[tid] += r[tid + s];
        __syncthreads();
    }
    if (tid == 0) out[0] = r[0] / (float)B_;
}

extern "C" void kernel_launch(void* const* d_in, const int* in_sizes, int n_in,
                              void* d_out, int out_size, void* d_ws, size_t ws_size,
                              hipStream_t stream) {
    (void)in_sizes; (void)n_in; (void)out_size; (void)ws_size;
    const float* src_onehot  = (const float*)d_in[0];
    const float* qin         = (const float*)d_in[1];
    const float* predictions = (const float*)d_in[2];
    const int*   tgt         = (const int*)  d_in[3];
    // d_in[4] = tgt_onehot: redundant given tgt (one_hot(tgt)); not read.
    const float* input_mask  = (const float*)d_in[5];
    const int*   timesteps   = (const int*)  d_in[6];
    const float* Qmat        = (const float*)d_in[7];
    const float* Qbar        = (const float*)d_in[8];

    float* partLoss = (float*)d_ws;                 // B*NBLK floats
    float* partMask = partLoss + (B_ * NBLK);       // B*NBLK floats

    d3pm_stage1<<<dim3(NBLK, B_), dim3(256), 0, stream>>>(
        src_onehot, qin, predictions, tgt, input_mask, timesteps, Qmat, Qbar,
        partLoss, partMask);
    d3pm_stage2<<<dim3(1), dim3(64), 0, stream>>>(partLoss, partMask, (float*)d_out);
}